// GNN_27393301413931
// MI455X (gfx1250) — compile-verified
//
#include <hip/hip_runtime.h>

typedef float v2f __attribute__((ext_vector_type(2)));
typedef float v8f __attribute__((ext_vector_type(8)));

// ---------------- utility fills ----------------
__global__ void fill_f(float* __restrict__ p, int n) {
    int i = blockIdx.x * blockDim.x + threadIdx.x;
    if (i < n) p[i] = 0.0f;
}
__global__ void fill_i(int* __restrict__ p, int n) {
    int i = blockIdx.x * blockDim.x + threadIdx.x;
    if (i < n) p[i] = 0;
}

// ---------------- graph preprocessing ----------------
// deg[i] = sum of edge weights into i ; cnt[i] = #edges into i
__global__ void hist_kernel(const int* __restrict__ dst, const float* __restrict__ ew,
                            float* __restrict__ deg, int* __restrict__ cnt, int E) {
    int e = blockIdx.x * blockDim.x + threadIdx.x;
    if (e < E) {
        int d = dst[e];
        atomicAdd(&deg[d], ew[e]);
        atomicAdd(&cnt[d], 1);
    }
}

// in-place: dis = rsqrt(deg + 1)  (self-loop weight 1.0, always > 0)
__global__ void finalize_dis(float* __restrict__ deg_dis, int n) {
    int i = blockIdx.x * blockDim.x + threadIdx.x;
    if (i < n) deg_dis[i] = rsqrtf(deg_dis[i] + 1.0f);
}

__global__ void norm_kernel(const int* __restrict__ src, const int* __restrict__ dst,
                            const float* __restrict__ ew, const float* __restrict__ dis,
                            float* __restrict__ nrm, int E) {
    int e = blockIdx.x * blockDim.x + threadIdx.x;
    if (e < E) nrm[e] = dis[src[e]] * ew[e] * dis[dst[e]];
}

// single-block exclusive scan over n counts; writes off[0..n], off[n] = total
__global__ void scan_kernel(const int* __restrict__ cnt, int* __restrict__ off, int n) {
    __shared__ int buf[1024];
    __shared__ int carry;
    if (threadIdx.x == 0) carry = 0;
    __syncthreads();
    for (int base = 0; base < n; base += 1024) {
        int i = base + (int)threadIdx.x;
        int v = (i < n) ? cnt[i] : 0;
        buf[threadIdx.x] = v;
        __syncthreads();
        for (int s = 1; s < 1024; s <<= 1) {
            int t = (threadIdx.x >= (unsigned)s) ? buf[threadIdx.x - s] : 0;
            __syncthreads();
            buf[threadIdx.x] += t;
            __syncthreads();
        }
        if (i < n) off[i] = carry + buf[threadIdx.x] - v;   // exclusive
        __syncthreads();
        if (threadIdx.x == 0) carry += buf[1023];
        __syncthreads();
    }
    if (threadIdx.x == 0) off[n] = carry;
}

// stable-ish permutation of edge ids grouped by dst (CSR column data)
__global__ void build_perm(const int* __restrict__ dst, const int* __restrict__ off,
                           int* __restrict__ cur, int* __restrict__ perm, int E) {
    int e = blockIdx.x * blockDim.x + threadIdx.x;
    if (e < E) {
        int d = dst[e];
        int p = off[d] + atomicAdd(&cur[d], 1);
        perm[p] = e;
    }
}

// ---------------- fp32 WMMA GEMM: T[N x FOUT] = X[N x FIN] @ W[FIN x FOUT] ----------------
// One wave computes a 16x16 tile of T, stepping K by 4 with V_WMMA_F32_16X16X4_F32.
// Requires: rows % 16 == 0, FOUT % 128 == 0 (grid exact -> EXEC all ones for WMMA).
template <int FIN, int FOUT>
__global__ void gemm_wmma(const float* __restrict__ X, const float* __restrict__ W,
                          float* __restrict__ T) {
    const int lane  = threadIdx.x & 31;
    const int wave  = threadIdx.x >> 5;                 // 8 waves / block
    const int row0  = blockIdx.x * 16;
    const int col0  = (blockIdx.y * 8 + wave) * 16;
    const int m     = lane & 15;                        // A row / D column index
    const int khalf = (lane >> 4) * 2;                  // 0 for lanes 0-15, 2 for 16-31

    const float* __restrict__ xrow = X + (size_t)(row0 + m) * FIN;
    const float* __restrict__ wcol = W + (size_t)(col0 + m);

    v8f acc = {};
#pragma unroll 4
    for (int k = 0; k < FIN; k += 4) {
        v2f a, b;
        // A 16x4 layout: lane(0-15) v0=K0,v1=K1 ; lane(16-31) v0=K2,v1=K3
        a.x = xrow[k + khalf];
        a.y = xrow[k + khalf + 1];
        // B 4x16 (row k striped across lanes, symmetric split on lane>>4)
        b.x = wcol[(size_t)(k + khalf) * FOUT];
        b.y = wcol[(size_t)(k + khalf + 1) * FOUT];
        acc = __builtin_amdgcn_wmma_f32_16x16x4_f32(false, a, false, b,
                                                    (short)0, acc, false, false);
    }
    // C/D layout: VGPR j -> row (j + 8*(lane>>4)), col (lane&15)
    const int rhalf = (lane >> 4) * 8;
#pragma unroll
    for (int j = 0; j < 8; ++j)
        T[(size_t)(row0 + rhalf + j) * FOUT + col0 + m] = acc[j];
}

// ---------------- fused GCN aggregation (gather form, no atomics) ----------------
// out[i,:] = act( t[i,:]*dis[i]^2 + sum_{e in CSR(i)} t[src[e],:]*norm[e] + bias )
template <int F, bool RELU>
__global__ void aggregate_gather(const int* __restrict__ off, const int* __restrict__ perm,
                                 const int* __restrict__ src, const float* __restrict__ nrm,
                                 const float* __restrict__ t, const float* __restrict__ dis,
                                 const float* __restrict__ bias, float* __restrict__ out,
                                 int n) {
    constexpr int F4 = F / 4;
    int idx = blockIdx.x * blockDim.x + threadIdx.x;
    int node = idx / F4;                 // F4 >= 32 -> whole wave shares one node
    if (node >= n) return;
    int f4 = idx - node * F4;

    float d  = dis[node];
    float d2 = d * d;
    float4 acc = ((const float4*)(t + (size_t)node * F))[f4];
    acc.x *= d2; acc.y *= d2; acc.z *= d2; acc.w *= d2;

    int jb = off[node], je = off[node + 1];
    for (int j = jb; j < je; ++j) {
        int   e = perm[j];               // uniform across the wave -> scalar-cached
        float w = nrm[e];
        float4 v = ((const float4*)(t + (size_t)src[e] * F))[f4];  // coalesced row read
        acc.x += v.x * w; acc.y += v.y * w; acc.z += v.z * w; acc.w += v.w * w;
    }
    float4 bb = ((const float4*)bias)[f4];
    acc.x += bb.x; acc.y += bb.y; acc.z += bb.z; acc.w += bb.w;
    if (RELU) {
        acc.x = fmaxf(acc.x, 0.0f); acc.y = fmaxf(acc.y, 0.0f);
        acc.z = fmaxf(acc.z, 0.0f); acc.w = fmaxf(acc.w, 0.0f);
    }
    ((float4*)(out + (size_t)node * F))[f4] = acc;
}

// ---------------- output head: t3[i] = h2[i,:] . W3  (FOUT = 1) ----------------
__global__ void gemm_out(const float* __restrict__ H, const float* __restrict__ W3,
                         float* __restrict__ T, int n, int fin) {
    int node = blockIdx.x * (blockDim.x >> 5) + (threadIdx.x >> 5);
    int lane = threadIdx.x & 31;
    if (node >= n) return;
    const float* h = H + (size_t)node * fin;
    float s = 0.0f;
    for (int k = lane; k < fin; k += 32) s += h[k] * W3[k];
    for (int o = 16; o > 0; o >>= 1) s += __shfl_down(s, o, 32);
    if (lane == 0) T[node] = s;
}

__global__ void aggregate_f1(const int* __restrict__ off, const int* __restrict__ perm,
                             const int* __restrict__ src, const float* __restrict__ nrm,
                             const float* __restrict__ t, const float* __restrict__ dis,
                             const float* __restrict__ b3, float* __restrict__ out, int n) {
    int node = blockIdx.x * blockDim.x + threadIdx.x;
    if (node >= n) return;
    float d = dis[node];
    float acc = t[node] * d * d;
    int jb = off[node], je = off[node + 1];
    for (int j = jb; j < je; ++j) {
        int e = perm[j];
        acc += t[src[e]] * nrm[e];
    }
    out[node] = acc + b3[0];
}

// ---------------- launcher ----------------
extern "C" void kernel_launch(void* const* d_in, const int* in_sizes, int n_in,
                              void* d_out, int out_size, void* d_ws, size_t ws_size,
                              hipStream_t stream) {
    const float* x  = (const float*)d_in[0];
    const int*   ei = (const int*)d_in[1];
    const float* ew = (const float*)d_in[2];
    const float* W1 = (const float*)d_in[3];
    const float* b1 = (const float*)d_in[4];
    const float* W2 = (const float*)d_in[5];
    const float* b2 = (const float*)d_in[6];
    const float* W3 = (const float*)d_in[7];
    const float* b3 = (const float*)d_in[8];

    const int IN = 128, H1D = 512, H2D = 256;
    const int N = in_sizes[0] / IN;      // 50000 (divisible by 16)
    const int E = in_sizes[2];           // 800000
    const int* srcA = ei;                // edge_index row 0
    const int* dstA = ei + E;            // edge_index row 1

    // workspace carve-up (float units); total ~53.2M floats (~213 MB)
    float* ws   = (float*)d_ws;
    float* dis  = ws;                        // N  (deg -> dis in place)
    float* nrm  = ws + (1 << 16);            // E
    int*   cnt  = (int*)(ws + 1000000);      // N
    int*   off  = (int*)(ws + 1060000);      // N+1
    int*   cur  = (int*)(ws + 1120000);      // N
    int*   perm = (int*)(ws + 1180000);      // E
    float* bufA = ws + 2000000;              // N*512
    float* bufB = bufA + (size_t)N * H1D;    // N*512

    const int B = 256;
    auto cd = [](int a, int b) { return (a + b - 1) / b; };

    // graph preprocessing (shared by all 3 layers)
    fill_f<<<cd(N, B), B, 0, stream>>>(dis, N);
    fill_i<<<cd(N, B), B, 0, stream>>>(cnt, N);
    fill_i<<<cd(N, B), B, 0, stream>>>(cur, N);
    hist_kernel<<<cd(E, B), B, 0, stream>>>(dstA, ew, dis, cnt, E);
    finalize_dis<<<cd(N, B), B, 0, stream>>>(dis, N);
    norm_kernel<<<cd(E, B), B, 0, stream>>>(srcA, dstA, ew, dis, nrm, E);
    scan_kernel<<<1, 1024, 0, stream>>>(cnt, off, N);
    build_perm<<<cd(E, B), B, 0, stream>>>(dstA, off, cur, perm, E);

    // Layer 1: t1 = x @ W1 ; h1 = relu(aggregate + b1)
    gemm_wmma<128, 512><<<dim3(N / 16, 512 / 128), 256, 0, stream>>>(x, W1, bufA);
    aggregate_gather<512, true><<<cd(N * (512 / 4), B), B, 0, stream>>>(
        off, perm, srcA, nrm, bufA, dis, b1, bufB, N);

    // Layer 2: t2 = h1 @ W2 ; h2 = relu(aggregate + b2)
    gemm_wmma<512, 256><<<dim3(N / 16, 256 / 128), 256, 0, stream>>>(bufB, W2, bufA);
    aggregate_gather<256, true><<<cd(N * (256 / 4), B), B, 0, stream>>>(
        off, perm, srcA, nrm, bufA, dis, b2, bufB, N);

    // Layer 3: t3 = h2 @ W3 (Fout=1) ; out = aggregate + b3 (no relu)
    gemm_out<<<cd(N, 8), 256, 0, stream>>>(bufB, W3, bufA, N, H2D);
    aggregate_f1<<<cd(N, B), B, 0, stream>>>(off, perm, srcA, nrm, bufA, dis, b3,
                                             (float*)d_out, N);
}